// GNNModel_19215683682694
// MI455X (gfx1250) — compile-verified
//
#include <hip/hip_runtime.h>

typedef __attribute__((ext_vector_type(2))) float v2f;
typedef __attribute__((ext_vector_type(8))) float v8f;

// ---------------------------------------------------------------------------
// Degree / normalization
// ---------------------------------------------------------------------------
__global__ void init_deg_kernel(float* __restrict__ deg, int n) {
    int i = blockIdx.x * blockDim.x + threadIdx.x;
    if (i < n) deg[i] = 1.0f;  // self-loop contributes 1 to every node's degree
}

__global__ void accum_deg_kernel(const int* __restrict__ dst,
                                 float* __restrict__ deg, int ne) {
    int i = blockIdx.x * blockDim.x + threadIdx.x;
    if (i < ne) atomicAdd(&deg[dst[i]], 1.0f);
}

__global__ void deg_to_dinv_kernel(float* __restrict__ d, int n) {
    int i = blockIdx.x * blockDim.x + threadIdx.x;
    if (i < n) d[i] = rsqrtf(d[i]);  // deg >= 1 always (self-loop)
}

// ---------------------------------------------------------------------------
// Zero-padded activation copy: x [NN,18] -> xp [NN_pad,20]
// ---------------------------------------------------------------------------
__global__ void pad_x_kernel(const float* __restrict__ x, float* __restrict__ xp,
                             int nn, int nn_pad) {
    int i = blockIdx.x * blockDim.x + threadIdx.x;
    int total = nn_pad * 20;
    if (i >= total) return;
    int n = i / 20;
    int k = i % 20;
    xp[i] = (n < nn && k < 18) ? x[n * 18 + k] : 0.0f;
}

// ---------------------------------------------------------------------------
// Weight repack: W [kd_true, nf] row-major -> Wp [kp/2][nf][2] K-pair
// interleaved + zero K-padding.  A lane's WMMA B-fragment
// B[k][n], B[k+1][n] (k even) then sits at consecutive addresses
// -> one 8B-aligned global_load_b64 per fragment.
// ---------------------------------------------------------------------------
__global__ void repack_w_kernel(const float* __restrict__ w, float* __restrict__ wp,
                                int kd_true, int kp, int nf) {
    int o = blockIdx.x * blockDim.x + threadIdx.x;  // kp*nf outputs
    if (o >= kp * nf) return;
    int j = o & 1;
    int n = (o >> 1) % nf;
    int p = (o >> 1) / nf;
    int k = 2 * p + j;
    wp[o] = (k < kd_true) ? w[k * nf + n] : 0.0f;
}

// ---------------------------------------------------------------------------
// GEMM via V_WMMA_F32_16X16X4_F32, fused epilogue: scale rows by dinv[m],
// write hs (scaled messages) and seed agg (self-loop init).
// One wave per 16-row M tile, all NT 16-col N tiles in registers.
// KD multiple of 4, M fully padded, W pair-packed -> guard-free b64 loads.
// ---------------------------------------------------------------------------
template <int KD, int NF, int NT>
__global__ void gemm_scale_kernel(const float* __restrict__ X,
                                  const float* __restrict__ Wp,  // pair-packed
                                  const float* __restrict__ dinv,
                                  float* __restrict__ hs,
                                  float* __restrict__ agg,
                                  int mtiles) {
    int wave = blockIdx.x * (blockDim.x >> 5) + (threadIdx.x >> 5);
    if (wave >= mtiles) return;          // uniform per-wave exit: EXEC stays all-1
    int lane = threadIdx.x & 31;
    int r16  = lane & 15;                // M index (A frag) / N index (B frag)
    int kh   = lane >> 4;                // K-half selector
    int mbase = wave * 16;
    int arow  = mbase + r16;

    v8f acc[NT] = {};

    static_assert(KD % 4 == 0, "KD must be multiple of 4");
    constexpr int KSTEPS = KD / 4;
#pragma unroll
    for (int ks = 0; ks < KSTEPS; ++ks) {
        int k0 = ks * 4 + kh * 2;
        // (arow*KD + k0) even, X >=8B aligned -> single b64 load
        v2f a = *(const v2f*)&X[arow * KD + k0];
        int p = ks * 2 + kh;             // K-pair index
#pragma unroll
        for (int t = 0; t < NT; ++t) {
            // pair-packed: both K rows of the fragment are adjacent -> b64 load
            v2f b = *(const v2f*)&Wp[(p * NF + t * 16 + r16) * 2];
            acc[t] = __builtin_amdgcn_wmma_f32_16x16x4_f32(
                false, a, false, b, (short)0, acc[t], false, false);
        }
    }

    // Epilogue: D[v + 8*kh][r16] lives in acc[t][v]; scale by dinv of its row.
    // Rows needed by this lane are the contiguous 8: [mbase+8*kh, +8) (32B aligned).
    const float4 dv0 = *(const float4*)&dinv[mbase + 8 * kh];
    const float4 dv1 = *(const float4*)&dinv[mbase + 8 * kh + 4];
    const float dv[8] = {dv0.x, dv0.y, dv0.z, dv0.w, dv1.x, dv1.y, dv1.z, dv1.w};

#pragma unroll
    for (int v = 0; v < 8; ++v) {
        int m = mbase + v + 8 * kh;
#pragma unroll
        for (int t = 0; t < NT; ++t) {
            float val = acc[t][v] * dv[v];
            int idx = m * NF + t * 16 + r16;
            hs[idx]  = val;   // scaled message buffer
            agg[idx] = val;   // accumulator seeded with self-loop term
        }
    }
}

// ---------------------------------------------------------------------------
// Edge scatter: agg[dst] += hs[src].  b128 gather + 4 no-return f32 atomics.
// Each thread moves one float4 (16B) of one edge's message.
// ---------------------------------------------------------------------------
template <int F4>  // F4 = feature_dim / 4
__global__ void scatter_kernel(const int* __restrict__ src,
                               const int* __restrict__ dst,
                               const float* __restrict__ hs,
                               float* __restrict__ agg, int ne) {
    unsigned int tid = blockIdx.x * blockDim.x + threadIdx.x;
    unsigned int total = (unsigned int)ne * F4;
    if (tid >= total) return;
    int e  = (int)(tid / F4);
    int fo = (int)(tid % F4) * 4;
    constexpr int F = F4 * 4;
    int s = src[e];
    int d = dst[e];
    const float4 val = *(const float4*)&hs[s * F + fo];
    float* p = &agg[d * F + fo];
    atomicAdd(p + 0, val.x);
    atomicAdd(p + 1, val.y);
    atomicAdd(p + 2, val.z);
    atomicAdd(p + 3, val.w);
}

// ---------------------------------------------------------------------------
// Finalize: out = dinv[n] * agg + b  (optionally ReLU), float4 vectorized.
// ---------------------------------------------------------------------------
template <int F, bool RELU>
__global__ void finalize_kernel(const float* __restrict__ agg,
                                const float* __restrict__ dinv,
                                const float* __restrict__ bias,
                                float* __restrict__ out, int nrows) {
    int i = blockIdx.x * blockDim.x + threadIdx.x;   // one float4 per thread
    constexpr int F4 = F / 4;
    if (i >= nrows * F4) return;
    int n  = i / F4;
    int fo = (i % F4) * 4;
    float dvn = dinv[n];
    const float4 a = *(const float4*)&agg[n * F + fo];
    const float4 b = *(const float4*)&bias[fo];
    float4 v;
    v.x = dvn * a.x + b.x;
    v.y = dvn * a.y + b.y;
    v.z = dvn * a.z + b.z;
    v.w = dvn * a.w + b.w;
    if (RELU) {
        v.x = fmaxf(v.x, 0.0f);
        v.y = fmaxf(v.y, 0.0f);
        v.z = fmaxf(v.z, 0.0f);
        v.w = fmaxf(v.w, 0.0f);
    }
    *(float4*)&out[n * F + fo] = v;
}

// ---------------------------------------------------------------------------
// Host orchestration
// ---------------------------------------------------------------------------
extern "C" void kernel_launch(void* const* d_in, const int* in_sizes, int n_in,
                              void* d_out, int out_size, void* d_ws, size_t ws_size,
                              hipStream_t stream) {
    const float* x  = (const float*)d_in[0];
    const int*   ei = (const int*)d_in[1];
    const float* W1 = (const float*)d_in[2];
    const float* b1 = (const float*)d_in[3];
    const float* W2 = (const float*)d_in[4];
    const float* b2 = (const float*)d_in[5];
    float* out = (float*)d_out;

    const int NN = in_sizes[0] / 18;          // 100000
    const int NE = in_sizes[1] / 2;           // 3200000
    const int NN_pad = (NN + 15) & ~15;       // multiple of 16 (== NN here)
    const int* src = ei;
    const int* dst = ei + NE;

    // Workspace carve-out (256B-aligned regions)
    char* ws = (char*)d_ws;
    size_t off = 0;
    auto carve = [&](size_t bytes) -> char* {
        char* p = ws + off;
        off += (bytes + 255) & ~(size_t)255;
        return p;
    };
    float* dinv = (float*)carve((size_t)NN_pad * 4);
    float* xp   = (float*)carve((size_t)NN_pad * 20 * 4);
    float* W1p  = (float*)carve((size_t)20 * 64 * 4);   // pair-packed, K-padded
    float* W2p  = (float*)carve((size_t)64 * 32 * 4);   // pair-packed
    float* hs1  = (float*)carve((size_t)NN_pad * 64 * 4);
    float* agg1 = (float*)carve((size_t)NN_pad * 64 * 4);
    float* h1   = (float*)carve((size_t)NN_pad * 64 * 4);
    float* hs2  = hs1;   // layer-1 buffers dead after finalize1 -> reuse
    float* agg2 = agg1;

    const int B = 256;

    // --- normalization coefficients (padded rows get dinv = 1, harmless) ---
    init_deg_kernel<<<(NN_pad + B - 1) / B, B, 0, stream>>>(dinv, NN_pad);
    accum_deg_kernel<<<(NE + B - 1) / B, B, 0, stream>>>(dst, dinv, NE);
    deg_to_dinv_kernel<<<(NN_pad + B - 1) / B, B, 0, stream>>>(dinv, NN_pad);

    // --- padded / pair-packed operands ---
    pad_x_kernel<<<(NN_pad * 20 + B - 1) / B, B, 0, stream>>>(x, xp, NN, NN_pad);
    repack_w_kernel<<<(20 * 64 + B - 1) / B, B, 0, stream>>>(W1, W1p, 18, 20, 64);
    repack_w_kernel<<<(64 * 32 + B - 1) / B, B, 0, stream>>>(W2, W2p, 64, 64, 32);

    const int mtiles = NN_pad / 16;           // 6250
    const int gblk = (mtiles + 3) / 4;        // 4 waves / block

    // --- layer 1: [NN,20] @ [20,64], ReLU ---
    gemm_scale_kernel<20, 64, 4><<<gblk, 128, 0, stream>>>(xp, W1p, dinv, hs1, agg1, mtiles);
    {
        unsigned int total = (unsigned int)NE * 16u;
        scatter_kernel<16><<<(total + B - 1) / B, B, 0, stream>>>(src, dst, hs1, agg1, NE);
    }
    finalize_kernel<64, true><<<(NN_pad * 16 + B - 1) / B, B, 0, stream>>>(agg1, dinv, b1, h1, NN_pad);

    // --- layer 2: [NN,64] @ [64,32] ---
    gemm_scale_kernel<64, 32, 2><<<gblk, 128, 0, stream>>>(h1, W2p, dinv, hs2, agg2, mtiles);
    {
        unsigned int total = (unsigned int)NE * 8u;
        scatter_kernel<8><<<(total + B - 1) / B, B, 0, stream>>>(src, dst, hs2, agg2, NE);
    }
    // output writes bounded to the true NN rows
    finalize_kernel<32, false><<<(NN * 8 + B - 1) / B, B, 0, stream>>>(agg2, dinv, b2, out, NN);
}